// GCN_90366111908565
// MI455X (gfx1250) — compile-verified
//
#include <hip/hip_runtime.h>

#define N_NODES 50000
#define N_EDGES 800000
#define IN_F    256
#define HID     128
#define N_CLS   16

typedef float v2f __attribute__((ext_vector_type(2)));
typedef float v8f __attribute__((ext_vector_type(8)));

// ---------------- degree + normalization ----------------
__global__ void gcn_degree_kernel(const int* __restrict__ src, const int* __restrict__ dst,
                                  float* __restrict__ deg_out, float* __restrict__ deg_in) {
    int e = blockIdx.x * blockDim.x + threadIdx.x;
    if (e < N_EDGES) {
        atomicAdd(&deg_out[src[e]], 1.0f);
        atomicAdd(&deg_in[dst[e]],  1.0f);
    }
}

__global__ void gcn_invsqrt_kernel(float* __restrict__ deg_out, float* __restrict__ deg_in) {
    int i = blockIdx.x * blockDim.x + threadIdx.x;
    if (i < N_NODES) {
        deg_out[i] = rsqrtf(fmaxf(deg_out[i], 1.0f));
        deg_in[i]  = rsqrtf(fmaxf(deg_in[i],  1.0f));
    }
}

// ---------------- GEMM1: hpre = (feat * inv_out) @ W_in   [50000,256]x[256,128] ----------------
// 256 threads = 8 waves; block = one 16-row M tile; wave w = N tile w (8*16 = 128 cols).
__global__ __launch_bounds__(256) void gcn_gemm1_kernel(const float* __restrict__ feat,
                                                        const float* __restrict__ W,
                                                        const float* __restrict__ inv_out,
                                                        float* __restrict__ hpre) {
    const int wave = threadIdx.x >> 5;
    const int lane = threadIdx.x & 31;
    const int half = lane >> 4;     // 0: K pair {0,1}; 1: K pair {2,3}
    const int l    = lane & 15;
    const int m0   = blockIdx.x * 16;
    const int n0   = wave * 16;
    const int arow = m0 + l;
    const float s  = inv_out[arow];
    const float* ap = feat + (size_t)arow * IN_F + 2 * half;

    v8f acc = {};
    for (int k0 = 0; k0 < IN_F; k0 += 4) {
        const float2 av = *(const float2*)(ap + k0);
        v2f a; a.x = av.x * s; a.y = av.y * s;
        const int kb = k0 + 2 * half;
        v2f b; b.x = W[kb * HID + n0 + l]; b.y = W[(kb + 1) * HID + n0 + l];
        acc = __builtin_amdgcn_wmma_f32_16x16x4_f32(false, a, false, b, (short)0, acc,
                                                    false, false);
    }
#pragma unroll
    for (int r = 0; r < 8; ++r) {
        const int orow = m0 + r + 8 * half;
        hpre[(size_t)orow * HID + n0 + l] = acc[r];
    }
}

// ---------------- edge aggregation, HID-wide: msg[dst] += hpre[src] ----------------
// 32 threads per edge, each handles a float4 chunk (L2-resident gather + atomics).
__global__ void gcn_agg_hid_kernel(const int* __restrict__ src, const int* __restrict__ dst,
                                   const float* __restrict__ hpre, float* __restrict__ msg) {
    long long tid = (long long)blockIdx.x * blockDim.x + threadIdx.x;
    int e = (int)(tid >> 5);
    int q = ((int)tid & 31) << 2;
    if (e < N_EDGES) {
        const int s = src[e], d = dst[e];
        const float4 v = *(const float4*)(hpre + (size_t)s * HID + q);
        float* p = msg + (size_t)d * HID + q;
        atomicAdd(p + 0, v.x); atomicAdd(p + 1, v.y);
        atomicAdd(p + 2, v.z); atomicAdd(p + 3, v.w);
    }
}

// ---------------- h = relu(msg * inv_in + b_in) ----------------
__global__ void gcn_relu_hid_kernel(const float* __restrict__ msg, const float* __restrict__ inv_in,
                                    const float* __restrict__ b, float* __restrict__ h) {
    int tid = blockIdx.x * blockDim.x + threadIdx.x;
    if (tid < N_NODES * HID) {
        const int n = tid >> 7, f = tid & 127;
        h[tid] = fmaxf(msg[tid] * inv_in[n] + b[f], 0.0f);
    }
}

// ---------------- GEMM2/3 fused: p1/p2 = (h*inv_out) @ {W_c,W_s}  [50000,128]x[128,16] ----------------
// 8 waves/block, each wave one M tile; A fragment shared by two accumulators.
__global__ __launch_bounds__(256) void gcn_gemm2_kernel(const float* __restrict__ h,
                                                        const float* __restrict__ Wc,
                                                        const float* __restrict__ Ws,
                                                        const float* __restrict__ inv_out,
                                                        float* __restrict__ p1,
                                                        float* __restrict__ p2) {
    const int wave = threadIdx.x >> 5;
    const int lane = threadIdx.x & 31;
    const int half = lane >> 4;
    const int l    = lane & 15;
    int mtile = blockIdx.x * 8 + wave;
    if (mtile > (N_NODES / 16) - 1) mtile = (N_NODES / 16) - 1;  // wave-uniform clamp, EXEC stays all-1s
    const int m0   = mtile * 16;
    const int arow = m0 + l;
    const float s  = inv_out[arow];
    const float* ap = h + (size_t)arow * HID + 2 * half;

    v8f c1 = {}, c2 = {};
    for (int k0 = 0; k0 < HID; k0 += 4) {
        const float2 av = *(const float2*)(ap + k0);
        v2f a; a.x = av.x * s; a.y = av.y * s;
        const int kb = k0 + 2 * half;
        v2f b1; b1.x = Wc[kb * N_CLS + l]; b1.y = Wc[(kb + 1) * N_CLS + l];
        v2f b2; b2.x = Ws[kb * N_CLS + l]; b2.y = Ws[(kb + 1) * N_CLS + l];
        c1 = __builtin_amdgcn_wmma_f32_16x16x4_f32(false, a, false, b1, (short)0, c1, false, false);
        c2 = __builtin_amdgcn_wmma_f32_16x16x4_f32(false, a, false, b2, (short)0, c2, false, false);
    }
#pragma unroll
    for (int r = 0; r < 8; ++r) {
        const int orow = m0 + r + 8 * half;
        p1[(size_t)orow * N_CLS + l] = c1[r];
        p2[(size_t)orow * N_CLS + l] = c2[r];
    }
}

// ---------------- edge aggregation, N_CLS-wide, both branches ----------------
// 8 threads per edge: j<4 -> p1 chunk j, j>=4 -> p2 chunk j-4.
__global__ void gcn_agg_cls_kernel(const int* __restrict__ src, const int* __restrict__ dst,
                                   const float* __restrict__ p1, const float* __restrict__ p2,
                                   float* __restrict__ m1, float* __restrict__ m2) {
    long long tid = (long long)blockIdx.x * blockDim.x + threadIdx.x;
    int e = (int)(tid >> 3);
    int j = (int)tid & 7;
    if (e < N_EDGES) {
        const int s = src[e], d = dst[e];
        const float* sb = (j < 4) ? p1 : p2;
        float*       db = (j < 4) ? m1 : m2;
        const int q = (j & 3) << 2;
        const float4 v = *(const float4*)(sb + (size_t)s * N_CLS + q);
        float* p = db + (size_t)d * N_CLS + q;
        atomicAdd(p + 0, v.x); atomicAdd(p + 1, v.y);
        atomicAdd(p + 2, v.z); atomicAdd(p + 3, v.w);
    }
}

// ---------------- final: out = [relu(m1*inv_in+b_c) ; relu(m2*inv_in+b_s)] ----------------
__global__ void gcn_final_kernel(const float* __restrict__ m1, const float* __restrict__ m2,
                                 const float* __restrict__ inv_in, const float* __restrict__ bc,
                                 const float* __restrict__ bs, float* __restrict__ out) {
    int tid = blockIdx.x * blockDim.x + threadIdx.x;
    if (tid < N_NODES * N_CLS) {
        const int n = tid >> 4, c = tid & 15;
        const float is = inv_in[n];
        out[tid]                   = fmaxf(m1[tid] * is + bc[c], 0.0f);
        out[N_NODES * N_CLS + tid] = fmaxf(m2[tid] * is + bs[c], 0.0f);
    }
}

extern "C" void kernel_launch(void* const* d_in, const int* in_sizes, int n_in,
                              void* d_out, int out_size, void* d_ws, size_t ws_size,
                              hipStream_t stream) {
    const float* feat = (const float*)d_in[0];
    const int*   src  = (const int*)d_in[1];
    const int*   dst  = (const int*)d_in[2];
    const float* W_in = (const float*)d_in[3];
    const float* b_in = (const float*)d_in[4];
    const float* W_c  = (const float*)d_in[5];
    const float* b_c  = (const float*)d_in[6];
    const float* W_s  = (const float*)d_in[7];
    const float* b_s  = (const float*)d_in[8];
    float* out = (float*)d_out;
    float* ws  = (float*)d_ws;

    // Workspace layout (floats). Atomic-accumulated regions packed first for a single memset.
    float* deg_out = ws;                //     50,000  -> becomes inv_sqrt_out
    float* deg_in  = ws +    50000;     //     50,000  -> becomes inv_sqrt_in
    float* msg     = ws +   100000;     //  6,400,000
    float* m1      = ws +  6500000;     //    800,000
    float* m2      = ws +  7300000;     //    800,000
    float* hpre    = ws +  8100000;     //  6,400,000  (reused as h after relu)
    float* p1      = ws + 14500000;     //    800,000
    float* p2      = ws + 15300000;     //    800,000   (total 16.1M floats = 64.4 MB)

    hipMemsetAsync(ws, 0, 8100000ull * sizeof(float), stream);

    gcn_degree_kernel <<<(N_EDGES + 255) / 256, 256, 0, stream>>>(src, dst, deg_out, deg_in);
    gcn_invsqrt_kernel<<<(N_NODES + 255) / 256, 256, 0, stream>>>(deg_out, deg_in);

    gcn_gemm1_kernel  <<<N_NODES / 16, 256, 0, stream>>>(feat, W_in, deg_out, hpre);
    gcn_agg_hid_kernel<<<(N_EDGES * 32) / 256, 256, 0, stream>>>(src, dst, hpre, msg);
    gcn_relu_hid_kernel<<<(N_NODES * HID + 255) / 256, 256, 0, stream>>>(msg, deg_in, b_in, hpre);

    gcn_gemm2_kernel  <<<(N_NODES / 16 + 7) / 8, 256, 0, stream>>>(hpre, W_c, W_s, deg_out, p1, p2);
    gcn_agg_cls_kernel<<<(N_EDGES * 8) / 256, 256, 0, stream>>>(src, dst, p1, p2, m1, m2);

    gcn_final_kernel  <<<(N_NODES * N_CLS + 255) / 256, 256, 0, stream>>>(m1, m2, deg_in, b_c, b_s, out);
}